// DecodeBox_55628416418025
// MI455X (gfx1250) — compile-verified
//
#include <hip/hip_runtime.h>

// DecodeBox: [B,30,D,H,W] f32 -> [B, 3*D*H*W, 10] f32
// Pure streaming transform; HBM-roofline ~10.8us @ 23.3 TB/s.
// One thread = 4 consecutive spatial positions of one (b, anchor):
//   10x global_load_b128 (nontemporal) -> trans ops -> 10x global_store_b128 (nontemporal)

typedef float v4f __attribute__((ext_vector_type(4)));

#define K_W 64
#define K_H 64
#define K_D 64
#define K_S (K_W * K_H * K_D)   // 262144 spatial positions
#define K_S4 (K_S / 4)          // 65536 per-thread work units per (b,a)
#define K_A 3                   // anchors
#define K_ATTR 10               // 5 + NUM_CLASSES

__device__ __forceinline__ float fast_sigmoid(float x) {
    // 1/(1+exp(-x)) ; exp(-x) = exp2(-x*log2(e)) -> v_exp_f32, v_rcp_f32
    float e = __builtin_amdgcn_exp2f(x * -1.44269504088896340736f);
    return __builtin_amdgcn_rcpf(1.0f + e);
}

__global__ __launch_bounds__(256) void decode_box_kernel(
    const float* __restrict__ in, float* __restrict__ out, int nthreads) {

    int tid = blockIdx.x * blockDim.x + threadIdx.x;
    if (tid >= nthreads) return;

    // tid -> (b, a, s4); S4 = 2^16 so use shifts
    int s4 = tid & (K_S4 - 1);
    int ba = tid >> 16;
    int a  = ba % K_A;
    int b  = ba / K_A;

    int s0 = s4 << 2;                       // first of 4 spatial positions
    float gx = (float)(s0 & (K_W - 1));     // w index (w..w+3 within same row)
    float gy = (float)((s0 >> 6) & (K_H - 1));
    float gz = (float)(s0 >> 12);

    // anchor width: ANCHORS[a][0]; stride factors cancel in bl = exp(l)*ANCHORS[a][0]
    float aw = (a == 0) ? 10.0f : ((a == 1) ? 16.0f : 33.0f);

    // ---- load 10 attribute planes, 4 consecutive elements each (b128, NT) ----
    const float* inp = in + ((size_t)(b * (K_A * K_ATTR) + a * K_ATTR)) * K_S + s0;
    v4f p[K_ATTR];
#pragma unroll
    for (int j = 0; j < K_ATTR; ++j) {
        p[j] = __builtin_nontemporal_load((const v4f*)(inp + (size_t)j * K_S));
    }

    // ---- decode: r[k*10 + attr] for k = 0..3 ----
    const float LOG2E = 1.44269504088896340736f;
    float r[4 * K_ATTR];
#pragma unroll
    for (int k = 0; k < 4; ++k) {
        r[k * K_ATTR + 0] = (fast_sigmoid(p[0][k]) + gx + (float)k) * 4.0f; // bx
        r[k * K_ATTR + 1] = (fast_sigmoid(p[1][k]) + gy) * 4.0f;            // by
        r[k * K_ATTR + 2] = (fast_sigmoid(p[2][k]) + gz) * 4.0f;            // bz
        r[k * K_ATTR + 3] = __builtin_amdgcn_exp2f(p[3][k] * LOG2E) * aw;   // bl
        r[k * K_ATTR + 4] = fast_sigmoid(p[4][k]);                          // conf
#pragma unroll
        for (int j = 5; j < K_ATTR; ++j) {
            r[k * K_ATTR + j] = fast_sigmoid(p[j][k]);                      // classes
        }
    }

    // ---- store 40 contiguous floats = 10x b128 (NT), 16B-aligned ----
    float* op = out + ((size_t)(b * K_A + a) * K_S + s0) * K_ATTR;
#pragma unroll
    for (int c = 0; c < K_ATTR; ++c) {
        v4f v;
        v.x = r[4 * c + 0];
        v.y = r[4 * c + 1];
        v.z = r[4 * c + 2];
        v.w = r[4 * c + 3];
        __builtin_nontemporal_store(v, (v4f*)(op + 4 * c));
    }
}

extern "C" void kernel_launch(void* const* d_in, const int* in_sizes, int n_in,
                              void* d_out, int out_size, void* d_ws, size_t ws_size,
                              hipStream_t stream) {
    (void)n_in; (void)out_size; (void)d_ws; (void)ws_size;
    const float* in = (const float*)d_in[0];
    float* out = (float*)d_out;

    // each thread consumes 40 input elements (10 attrs x 4 positions)
    int nthreads = in_sizes[0] / 40;
    int block = 256;                       // 8 wave32 waves
    int grid = (nthreads + block - 1) / block;
    decode_box_kernel<<<grid, block, 0, stream>>>(in, out, nthreads);
}